// GlassBlur_2190433321202
// MI455X (gfx1250) — compile-verified
//
#include <hip/hip_runtime.h>

// GlassBlur (sigma=0.05 -> gaussian is identity): output = clip(permute(img)).
// The permutation comes from a sequential scan of 222x222 swaps, each swapping
// pixel (h,w) with (h+dy, w+dx), dx,dy in {-1,0}, h and w both descending.
// Each pixel is touched by at most 4 steps, so per-pixel backward provenance
// tracing is exact and embarrassingly parallel.

#define IMG_H 224
#define IMG_W 224
#define SROW  222             // steps per row (b = 223..2)
#define NWIN  16              // dxy rows staged in LDS per block
#define ROWB  (SROW * 2 * 4)  // bytes per step-row of dxy = 1776 = 111 * 16

typedef __attribute__((ext_vector_type(4))) int v4i;
typedef __attribute__((address_space(1))) v4i* global_v4i_p;
typedef __attribute__((address_space(3))) v4i* lds_v4i_p;
typedef __attribute__((address_space(3))) int* lds_i_p;

__global__ __launch_bounds__(IMG_W) void glass_blur_trace_kernel(
    const float* __restrict__ img,
    const int*   __restrict__ dxy,
    float*       __restrict__ out) {
  __shared__ __align__(16) int lds_dxy[NWIN * SROW * 2];  // 28416 bytes

  const int h = blockIdx.x;
  const int w = threadIdx.x;

  // Prefetch the starting pixel's cacheline; src is usually nearby.
  __builtin_prefetch(&img[(h * IMG_W + w) * 3], 0, 0);

  // Stage dxy window for step rows a in [amin, amin+nrows).
  const int amin  = (h < 2) ? 2 : h;
  int nrows = 224 - amin;
  if (nrows > NWIN) nrows = NWIN;
  const int nchunks = nrows * (ROWB / 16);  // 16B chunks

#if defined(__has_builtin) && __has_builtin(__builtin_amdgcn_global_load_async_to_lds_b128)
  for (int k = threadIdx.x; k < nchunks; k += blockDim.x) {
    const int r = k / (ROWB / 16);
    const int c = k - r * (ROWB / 16);
    const char* g = (const char*)dxy + (size_t)(223 - (amin + r)) * ROWB + (size_t)c * 16;
    char* l = (char*)lds_dxy + r * ROWB + c * 16;
    __builtin_amdgcn_global_load_async_to_lds_b128((global_v4i_p)g, (lds_v4i_p)l, 0, 0);
  }
  // Wait for this wave's async-to-LDS ops; also escapes the LDS pointer so the
  // compiler treats lds_dxy as written by the async intrinsic.
  asm volatile("s_wait_asynccnt 0" : : "v"(&lds_dxy[0]) : "memory");
#else
  for (int k = threadIdx.x; k < nchunks; k += blockDim.x) {
    const int r = k / (ROWB / 16);
    const int c = k - r * (ROWB / 16);
    const int4* g = (const int4*)((const char*)dxy + (size_t)(223 - (amin + r)) * ROWB + (size_t)c * 16);
    int4* l = (int4*)((char*)lds_dxy + r * ROWB + c * 16);
    *l = *g;
  }
#endif
  __syncthreads();

  // Backward provenance trace. Bound (la,lb): a step (a,b) is strictly earlier
  // in execution than the bound iff (a > la) || (a == la && b > lb).
  int ph = h, pw = w;
  int la = -1, lb = 0;  // initial bound: everything allowed

  for (;;) {
    int na = -1, nb = -1;
    // Candidate steps touching (ph,pw), latest-executed first:
    // (ph,pw), (ph,pw+1), (ph+1,pw), (ph+1,pw+1)
#pragma unroll
    for (int ci = 0; ci < 4; ++ci) {
      const int a = ph + (ci >> 1);
      const int b = pw + (ci & 1);
      if (a < 2 || a > 223 || b < 2 || b > 223) continue;
      if (!((a > la) || (a == la && b > lb))) continue;  // must be earlier than bound
      const int j = 223 - b;

      const int oc = a - amin;
      const bool in_win = (unsigned)oc < (unsigned)nrows;
      int2 v;
      if (in_win) {
        // Fast path: force a real ds_load_b64. If left to the compiler, this
        // load gets sunk/merged with the global fallback into a flat_load
        // (generic pointer), paying the FLAT aperture + dual-counter cost.
        const int o = (oc * SROW + j) * 2;
        const unsigned lds_off = (unsigned)(unsigned long long)(lds_i_p)&lds_dxy[o];
        asm volatile("ds_load_b64 %0, %1\n\t"
                     "s_wait_dscnt 0"
                     : "=v"(v) : "v"(lds_off));
      } else {
        const int t = ((223 - a) * SROW + j) * 2;
        v = *(const int2*)&dxy[t];
      }
      const int dx = v.x, dy = v.y;

      if ((dx | dy) == 0) continue;  // no-op swap
      const int qa = a + dy, qb = b + dx;  // partner
      if (a == ph && b == pw) {            // pos was the primary -> value came from partner
        na = qa; nb = qb; la = a; lb = b;
        break;
      }
      if (qa == ph && qb == pw) {          // pos was the partner -> value came from primary
        na = a; nb = b; la = a; lb = b;
        break;
      }
    }
    if (na < 0) break;
    ph = na; pw = nb;
  }

  const int src = (ph * IMG_W + pw) * 3;
  const int dst = (h * IMG_W + w) * 3;
  const float r0 = img[src + 0];
  const float r1 = img[src + 1];
  const float r2 = img[src + 2];
  out[dst + 0] = fminf(fmaxf(r0, 0.0f), 1.0f);
  out[dst + 1] = fminf(fmaxf(r1, 0.0f), 1.0f);
  out[dst + 2] = fminf(fmaxf(r2, 0.0f), 1.0f);
}

extern "C" void kernel_launch(void* const* d_in, const int* in_sizes, int n_in,
                              void* d_out, int out_size, void* d_ws, size_t ws_size,
                              hipStream_t stream) {
  (void)in_sizes; (void)n_in; (void)out_size; (void)d_ws; (void)ws_size;
  const float* img = (const float*)d_in[0];
  const int*   dxy = (const int*)d_in[1];
  float*       out = (float*)d_out;
  dim3 grid(IMG_H), block(IMG_W);
  glass_blur_trace_kernel<<<grid, block, 0, stream>>>(img, dxy, out);
}